// Linear3D_50835232915724
// MI455X (gfx1250) — compile-verified
//
#include <hip/hip_runtime.h>
#include <hip/hip_bf16.h>

// Grouped-GEMM MoE gather: out[b,:] = x[b,:] @ W[sel[b]] + bias[sel[b]]
// Uses CDNA5 V_WMMA_F32_16X16X4_F32 (full f32 precision, matches reference).

#define B_SZ    1024
#define IN_SZ   512
#define OUT_SZ  512
#define OPT_SZ  16
#define MAX_TILES 64          // ceil(B_SZ/16)
#define LDSK    516           // 512 + 4 pad: 516%64==4 -> conflict-free A-frag reads

typedef __attribute__((ext_vector_type(2))) float v2f;
typedef __attribute__((ext_vector_type(4))) float v4f;
typedef __attribute__((ext_vector_type(8))) float v8f;

// ---------------- pass 0: zero per-expert counters ----------------
__global__ void k_zero(int* __restrict__ cnt) {
    if (threadIdx.x < OPT_SZ) cnt[threadIdx.x] = 0;
}

// ---------------- pass 1: scatter rows into per-expert lists ----------------
__global__ void k_scatter(const int* __restrict__ sel,
                          int* __restrict__ cnt,
                          int* __restrict__ rowlist) {
    int b = blockIdx.x * blockDim.x + threadIdx.x;
    if (b < B_SZ) {
        int e   = sel[b];
        int pos = atomicAdd(&cnt[e], 1);
        rowlist[e * B_SZ + pos] = b;
    }
}

// ---------------- pass 2: grouped GEMM via f32 WMMA ----------------
// Block: 128 threads (4 waves). One block = (expert e, 16-row tile, 128-col slab).
// Each wave owns two 16x16 output tiles (reuses its A fragment across both).
__global__ __launch_bounds__(128) void k_gemm(
        const float* __restrict__ x,       // [B, IN]
        const float* __restrict__ w,       // [OPT, IN, OUT]
        const float* __restrict__ bias,    // [OPT, OUT]
        const int*   __restrict__ cnt,     // [OPT]
        const int*   __restrict__ rowlist, // [OPT, B]
        float*       __restrict__ out) {   // [B, OUT]

    __shared__ float lds_x[16 * LDSK];     // 16-row x tile, padded stride
    __shared__ int   s_rows[16];

    const int e    = blockIdx.x >> 6;      // / MAX_TILES
    const int tile = blockIdx.x & (MAX_TILES - 1);
    const int c    = cnt[e];
    if (tile * 16 >= c) return;            // block-uniform: EXEC stays all-1s below

    const int tid = threadIdx.x;
    if (tid < 16) {
        int gr = tile * 16 + tid;
        s_rows[tid] = (gr < c) ? rowlist[e * B_SZ + gr] : -1;
    }
    __syncthreads();

    // Stage x tile -> LDS. Per j: 128 threads load one row's 512 floats as
    // float4 (fully coalesced 2KB burst); invalid (padding) rows get zeros.
    for (int j = 0; j < 16; ++j) {
        int row = s_rows[j];
        v4f val = (v4f){0.f, 0.f, 0.f, 0.f};
        if (row >= 0) val = *(const v4f*)(x + row * IN_SZ + tid * 4);
        *(v4f*)(lds_x + j * LDSK + tid * 4) = val;
    }
    __syncthreads();

    const int lane = tid & 31;
    const int wave = tid >> 5;
    const int nlo  = lane & 15;            // N (and A-row M) sub-index
    const int koff = (lane >> 4) << 1;     // lanes 16-31 hold K+2/K+3
    const int n0   = blockIdx.y * 128 + wave * 32 + nlo;

    const float* wp   = w + (size_t)e * (IN_SZ * OUT_SZ);
    const float* aptr = lds_x + nlo * LDSK + koff;   // A-frag base (M = lane&15)

    // C init = bias[e][n] (constant per lane across all 8 acc rows)
    const float b0v = bias[e * OUT_SZ + n0];
    const float b1v = bias[e * OUT_SZ + n0 + 16];
    v8f acc0, acc1;
#pragma unroll
    for (int j = 0; j < 8; ++j) { acc0[j] = b0v; acc1[j] = b1v; }

#pragma unroll 4
    for (int k = 0; k < IN_SZ; k += 4) {
        // A fragment: x[M][k+koff], x[M][k+koff+1] via one ds_load_b64
        v2f a = *(const v2f*)(aptr + k);
        // B fragments: w[k+koff][n], w[k+koff+1][n] for the two N-tiles
        const float* wk = wp + (size_t)(k + koff) * OUT_SZ + n0;
        v2f bf0, bf1;
        bf0.x = wk[0];       bf0.y = wk[OUT_SZ];
        bf1.x = wk[16];      bf1.y = wk[OUT_SZ + 16];
        acc0 = __builtin_amdgcn_wmma_f32_16x16x4_f32(
                   false, a, false, bf0, (short)0, acc0, false, false);
        acc1 = __builtin_amdgcn_wmma_f32_16x16x4_f32(
                   false, a, false, bf1, (short)0, acc1, false, false);
    }

    // D layout: VGPR j -> M = j + 8*(lane>=16), N = lane&15
    const int mhi = (lane >> 4) << 3;
#pragma unroll
    for (int j = 0; j < 8; ++j) {
        int row = s_rows[j + mhi];
        if (row >= 0) {
            out[row * OUT_SZ + n0]      = acc0[j];
            out[row * OUT_SZ + n0 + 16] = acc1[j];
        }
    }
}

extern "C" void kernel_launch(void* const* d_in, const int* in_sizes, int n_in,
                              void* d_out, int out_size, void* d_ws, size_t ws_size,
                              hipStream_t stream) {
    const float* x    = (const float*)d_in[0];
    const float* w    = (const float*)d_in[1];
    const float* bias = (const float*)d_in[2];
    const int*   sel  = (const int*)d_in[3];
    float*       out  = (float*)d_out;

    int* cnt     = (int*)d_ws;         // 16 ints
    int* rowlist = cnt + OPT_SZ;       // OPT_SZ * B_SZ ints (~64 KB)

    k_zero<<<1, 32, 0, stream>>>(cnt);
    k_scatter<<<B_SZ / 256, 256, 0, stream>>>(sel, cnt, rowlist);

    dim3 grid(OPT_SZ * MAX_TILES, OUT_SZ / 128);   // (1024, 4); empty tiles exit early
    k_gemm<<<grid, 128, 0, stream>>>(x, w, bias, cnt, rowlist, out);
}